// PolyConv_7138235646045
// MI455X (gfx1250) — compile-verified
//
#include <hip/hip_runtime.h>

typedef float v2f __attribute__((ext_vector_type(2)));
typedef float v8f __attribute__((ext_vector_type(8)));

#define DF 32
#define TB 256

// ---------------- setup kernels ----------------

__global__ void __launch_bounds__(TB) k_zero(float* __restrict__ p, int n) {
  int i = blockIdx.x * TB + threadIdx.x;
  if (i < n) p[i] = 0.0f;
}

__global__ void __launch_bounds__(TB) k_count_deg(const int* __restrict__ row,
                                                  float* __restrict__ deg, int nE) {
  int e = blockIdx.x * TB + threadIdx.x;
  if (e < nE)
    __hip_atomic_fetch_add(&deg[row[e]], 1.0f, __ATOMIC_RELAXED, __HIP_MEMORY_SCOPE_AGENT);
}

__global__ void __launch_bounds__(TB) k_deg_to_dis(float* __restrict__ d, int n) {
  int i = blockIdx.x * TB + threadIdx.x;
  if (i < n) { float v = d[i]; d[i] = (v > 0.0f) ? rsqrtf(v) : 0.0f; }
}

__global__ void __launch_bounds__(TB) k_edge_w(const int* __restrict__ row,
                                               const int* __restrict__ col,
                                               const float* __restrict__ dis,
                                               float* __restrict__ w, int nE) {
  int e = blockIdx.x * TB + threadIdx.x;
  if (e < nE) w[e] = -dis[row[e]] * dis[col[e]];
}

__global__ void __launch_bounds__(TB) k_init(const float* __restrict__ h,
                                             const float* __restrict__ theta,
                                             float* __restrict__ out,
                                             float* __restrict__ b1, int n) {
  int i = blockIdx.x * TB + threadIdx.x;
  if (i < n) { float v = h[i]; out[i] = theta[0] * v; b1[i] = v; }
}

// ---------------- scatter: dst[row] += w * src[col], lane = feature ----------------
// One block stages 256 edges of (row, col, w) into LDS via async-to-LDS, then
// each of the 8 waves processes 32 edges: per edge one coalesced 128B gather and
// one coalesced non-returning global_atomic_add_f32 burst (full cacheline at L2).

__global__ void __launch_bounds__(TB) k_scatter(const int* __restrict__ row,
                                                const int* __restrict__ col,
                                                const float* __restrict__ w,
                                                const float* __restrict__ src,
                                                float* __restrict__ dst, int nE) {
  __shared__ int   s_row[TB];
  __shared__ int   s_col[TB];
  __shared__ float s_w[TB];

  const int tid  = threadIdx.x;
  const int base = blockIdx.x * TB;
  int nval = nE - base;
  if (nval > TB) nval = TB;

#if __has_builtin(__builtin_amdgcn_global_load_async_to_lds_b32)
  // Builtin parameter types (from clang diagnostic): AS(1) int* src, AS(3) int* dst.
  typedef __attribute__((address_space(1))) int gint_t;
  typedef __attribute__((address_space(3))) int lint_t;
  if (tid < nval) {
    __builtin_amdgcn_global_load_async_to_lds_b32((gint_t*)(row + base + tid),
                                                  (lint_t*)&s_row[tid], 0, 0);
    __builtin_amdgcn_global_load_async_to_lds_b32((gint_t*)(col + base + tid),
                                                  (lint_t*)&s_col[tid], 0, 0);
    __builtin_amdgcn_global_load_async_to_lds_b32((gint_t*)(w + base + tid),
                                                  (lint_t*)&s_w[tid], 0, 0);
  }
#if __has_builtin(__builtin_amdgcn_s_wait_asynccnt)
  __builtin_amdgcn_s_wait_asynccnt(0);
#else
  asm volatile("s_wait_asynccnt 0" ::: "memory");
#endif
#else
  if (tid < nval) {
    s_row[tid] = row[base + tid];
    s_col[tid] = col[base + tid];
    s_w[tid]   = w[base + tid];
  }
#endif
  __syncthreads();

  const int lane = tid & 31;
  const int wid  = tid >> 5;
  const int e0   = wid * 32;

#pragma unroll 4
  for (int j = 0; j < 32; ++j) {
    int e = e0 + j;
    if (e >= nval) break;
    int   r  = s_row[e];           // DS broadcast (wave-uniform)
    int   c  = s_col[e];
    float ww = s_w[e];
    float v  = src[(size_t)c * DF + lane];              // coalesced 128B gather
    __hip_atomic_fetch_add(&dst[(size_t)r * DF + lane], // packed fadd, no return
                           ww * v, __ATOMIC_RELAXED, __HIP_MEMORY_SCOPE_AGENT);
  }
}

// ---------------- combine: out += theta[k] * F (WMMA), other = F ----------------
// One wave per 16x16 tile. D = (theta * I16) * F + C as 4 chained
// v_wmma_f32_16x16x4_f32 over K-blocks. EXEC all ones (wave-uniform guard).

__global__ void __launch_bounds__(TB) k_combine(const float* __restrict__ F,
                                                float* __restrict__ other,
                                                float* __restrict__ out,
                                                const float* __restrict__ theta,
                                                int k, int doCopy, int nTiles) {
  const int waveId = blockIdx.x * (TB / 32) + (threadIdx.x >> 5);
  if (waveId >= nTiles) return;                 // wave-uniform: EXEC stays all-1s
  const int lane = threadIdx.x & 31;
  const int half = lane >> 4;                   // 0: lanes 0-15, 1: lanes 16-31
  const int l16  = lane & 15;

  const int colTiles = DF / 16;                 // 2
  const int m0 = (waveId / colTiles) * 16;      // node-row base
  const int n0 = (waveId % colTiles) * 16;      // feature-col base

  const float th = theta[k];

  // Load C (current out tile) in the documented 16x16 f32 C/D layout:
  // VGPR r: lanes 0-15 -> M=r, lanes 16-31 -> M=r+8, N = lane%16.
  v8f acc;
  const float* cBase = out + (size_t)(m0 + 8 * half) * DF + n0 + l16;
#pragma unroll
  for (int r = 0; r < 8; ++r) acc[r] = cBase[(size_t)r * DF];

#pragma unroll
  for (int j = 0; j < 4; ++j) {
    // A_j (16x4) = theta on diagonal rows 4j..4j+3.
    // Layout: VGPR0 = (M=l16, K=2*half), VGPR1 = (M=l16, K=1+2*half).
    v2f a;
    a.x = (l16 == 4 * j + 2 * half)     ? th : 0.0f;
    a.y = (l16 == 4 * j + 1 + 2 * half) ? th : 0.0f;
    // B_j (4x16) = rows 4j..4j+3 of F tile; row striped across lanes per VGPR.
    v2f b;
    const float* fb = F + (size_t)(m0 + 4 * j + 2 * half) * DF + n0 + l16;
    b.x = fb[0];
    b.y = fb[DF];
    acc = __builtin_amdgcn_wmma_f32_16x16x4_f32(false, a, false, b,
                                                (short)0, acc, false, false);
  }

  float* dBase = out + (size_t)(m0 + 8 * half) * DF + n0 + l16;
#pragma unroll
  for (int r = 0; r < 8; ++r) dBase[(size_t)r * DF] = acc[r];

  if (doCopy) {                                 // ping-pong init for next iteration
    const float* fr = F + (size_t)(m0 + 8 * half) * DF + n0 + l16;
    float* ow = other + (size_t)(m0 + 8 * half) * DF + n0 + l16;
#pragma unroll
    for (int r = 0; r < 8; ++r) ow[(size_t)r * DF] = fr[(size_t)r * DF];
  }
}

// ---------------- launch ----------------

extern "C" void kernel_launch(void* const* d_in, const int* in_sizes, int n_in,
                              void* d_out, int out_size, void* d_ws, size_t ws_size,
                              hipStream_t stream) {
  (void)n_in; (void)out_size; (void)ws_size;

  const float* h     = (const float*)d_in[0];
  const int*   ei    = (const int*)d_in[1];
  const float* theta = (const float*)d_in[2];
  float*       out   = (float*)d_out;

  const int E   = in_sizes[1] / 2;
  const int N   = in_sizes[0] / DF;
  const int* row = ei;          // edge_index[0]
  const int* col = ei + E;      // edge_index[1]

  // Workspace layout (all L2-resident): dis | w | B1 | B2
  char* ws = (char*)d_ws;
  size_t oDis = 0;
  size_t oW   = ((size_t)N * 4 + 4095) & ~(size_t)4095;
  size_t oB1  = oW + (((size_t)E * 4 + 4095) & ~(size_t)4095);
  size_t oB2  = oB1 + (((size_t)N * DF * 4 + 4095) & ~(size_t)4095);
  float* dis = (float*)(ws + oDis);
  float* w   = (float*)(ws + oW);
  float* B1  = (float*)(ws + oB1);
  float* B2  = (float*)(ws + oB2);

  const int gN  = (N + TB - 1) / TB;
  const int gE  = (E + TB - 1) / TB;
  const int gNF = (N * DF + TB - 1) / TB;
  const int nTiles = (N / 16) * (DF / 16);
  const int gC  = (nTiles + (TB / 32) - 1) / (TB / 32);

  // Degree -> dis = deg^{-1/2} (in place) -> edge weights (computed once).
  k_zero<<<gN, TB, 0, stream>>>(dis, N);
  k_count_deg<<<gE, TB, 0, stream>>>(row, dis, E);
  k_deg_to_dis<<<gN, TB, 0, stream>>>(dis, N);
  k_edge_w<<<gE, TB, 0, stream>>>(row, col, dis, w, E);

  // out = theta0*h ; B1 = h (diagonal term of L pre-seeded for iter 1).
  k_init<<<gNF, TB, 0, stream>>>(h, theta, out, B1, N * DF);

  // iter 1: B1 = L h = P1 ; out += th1*P1 ; B2 = P1
  k_scatter<<<gE, TB, 0, stream>>>(row, col, w, h, B1, E);
  k_combine<<<gC, TB, 0, stream>>>(B1, B2, out, theta, 1, 1, nTiles);
  // iter 2: B2 = L P1 = P2 ; out += th2*P2 ; B1 = P2
  k_scatter<<<gE, TB, 0, stream>>>(row, col, w, B1, B2, E);
  k_combine<<<gC, TB, 0, stream>>>(B2, B1, out, theta, 2, 1, nTiles);
  // iter 3: B1 = P3 ; out += th3*P3 ; B2 = P3
  k_scatter<<<gE, TB, 0, stream>>>(row, col, w, B2, B1, E);
  k_combine<<<gC, TB, 0, stream>>>(B1, B2, out, theta, 3, 1, nTiles);
  // iter 4: B2 = P4 ; out += th4*P4
  k_scatter<<<gE, TB, 0, stream>>>(row, col, w, B1, B2, E);
  k_combine<<<gC, TB, 0, stream>>>(B2, B1, out, theta, 4, 0, nTiles);
}